// MSDeformAttnTransformerEncoderOnly_fusionText_87316685128180
// MI455X (gfx1250) — compile-verified
//
#include <hip/hip_runtime.h>

// ---------------------------------------------------------------------------
// MSDeformAttn encoder (Deformable-DETR), MI455X / gfx1250, fp32 throughout.
//
// Roofline: ~240 GFLOP of GEMM over a working set that is fully L2-resident
// (192MB L2) -> L2-bandwidth / gather-latency bound, not FLOP bound.
// Therefore fp32 WMMA (V_WMMA_F32_16X16X4_F32) for exact-precision matrix
// math, and the effort goes into data movement:
//   * GEMM tiles staged global->LDS by the Tensor Data Mover
//     (TENSOR_LOAD_TO_LDS, TENSORcnt, double-buffered ping/pong so the DMA
//     of chunk k+1 overlaps the WMMAs of chunk k; pad_interval/pad_amount
//     reproduce the +1-dword row padding to dodge LDS bank conflicts;
//     tensor_dim1 = M - rowBase gives hardware zero-fill at the M edge)
//   * sampling kernel: one wave per (b,q,head), lane == head-channel
//     (HD == 32 == wave32) so every bilinear tap is one coalesced 128B gather
// ---------------------------------------------------------------------------

#define B_   2
#define D_   256
#define NH_  8
#define HD_  32
#define NLVL_ 4
#define NP_  4
#define DFF_ 1024
#define NLAYERS_ 6
#define Q_   13294            // 100*100 + 50*50 + 25*25 + 13*13
#define M_   (B_ * Q_)        // 26588 token rows

typedef __attribute__((ext_vector_type(2))) float v2f;
typedef __attribute__((ext_vector_type(8))) float v8f;
typedef __attribute__((ext_vector_type(4))) unsigned int u32x4;
typedef __attribute__((ext_vector_type(4))) int i32x4;
typedef __attribute__((ext_vector_type(8))) int i32x8;

#if defined(__has_builtin)
#if __has_builtin(__builtin_amdgcn_tensor_load_to_lds) && \
    __has_builtin(__builtin_amdgcn_s_wait_tensorcnt)
#define USE_TDM 1
#endif
#endif
#ifndef USE_TDM
#define USE_TDM 0
#endif

// ---------------------------------------------------------------------------
// 0) Flatten [B,D,H,W] level maps into token-major [B,Q,D]; pos += level_embed
// ---------------------------------------------------------------------------
__global__ __launch_bounds__(256) void flatten_kernel(
    const float* __restrict__ s0, const float* __restrict__ s1,
    const float* __restrict__ s2, const float* __restrict__ s3,
    const float* __restrict__ p0, const float* __restrict__ p1,
    const float* __restrict__ p2, const float* __restrict__ p3,
    const float* __restrict__ lev,
    float* __restrict__ x, float* __restrict__ pos)
{
    int idx = blockIdx.x * 256 + threadIdx.x;
    const int total = B_ * Q_ * D_;
    if (idx >= total) return;
    int d = idx & 255;
    int q = (idx >> 8) % Q_;
    int b = idx / (256 * Q_);

    const float* sp; const float* pp; int l, st, hw;
    if (q < 10000)      { l = 0; st = 0;     hw = 10000; sp = s0; pp = p0; }
    else if (q < 12500) { l = 1; st = 10000; hw = 2500;  sp = s1; pp = p1; }
    else if (q < 13125) { l = 2; st = 12500; hw = 625;   sp = s2; pp = p2; }
    else                { l = 3; st = 13125; hw = 169;   sp = s3; pp = p3; }
    int pq = q - st;
    size_t si = (size_t)(b * D_ + d) * hw + pq;   // [B,D,H*W]
    x[idx]   = sp[si];
    pos[idx] = pp[si] + lev[l * D_ + d];
}

// ---------------------------------------------------------------------------
// 1) Elementwise q = x + pos (float4)
// ---------------------------------------------------------------------------
__global__ __launch_bounds__(256) void add4_kernel(
    const float* __restrict__ a, const float* __restrict__ b,
    float* __restrict__ o, int n4)
{
    int i = blockIdx.x * 256 + threadIdx.x;
    if (i >= n4) return;
    float4 va = reinterpret_cast<const float4*>(a)[i];
    float4 vb = reinterpret_cast<const float4*>(b)[i];
    float4 vo = make_float4(va.x + vb.x, va.y + vb.y, va.z + vb.z, va.w + vb.w);
    reinterpret_cast<float4*>(o)[i] = vo;
}

// ---------------------------------------------------------------------------
// 2) WMMA fp32 GEMM:  Y[M,N] = A[M,K] @ W[K,N] + bias (opt ReLU)
//    256 threads = 8 waves; block tile 64x32; wave tile 16x16; K-chunk 32.
//    Tile staging via Tensor Data Mover, double buffered.
// ---------------------------------------------------------------------------
#define TM 64
#define TN 32
#define TK 32

#if USE_TDM
// Issue one 2D-tile TDM load: tile (tile_h rows) x (tile_w dwords), fp32,
// row stride `row_stride` elements, zero-fill for rows >= tensor_h.
// LDS rows are padded: after every 32 dwords stored, skip 1 dword
// (pad_interval code 4 == 32 dwords, pad_amount code 0 == 1 dword),
// matching the float[rows][33] LDS layout.
__device__ __forceinline__ void tdm_load_2d(
    unsigned int lds_byte, const float* gptr,
    unsigned int tile_w, unsigned int tile_h,
    unsigned int tensor_h, unsigned int row_stride)
{
    unsigned long long ga = (unsigned long long)(uintptr_t)gptr;
    u32x4 g0;
    g0[0] = 1u;                                   // count=1, user descriptor
    g0[1] = lds_byte;                             // LDS byte address
    g0[2] = (unsigned int)ga;                     // global addr [31:0]
    g0[3] = (unsigned int)((ga >> 32) & 0x01FFFFFFu) | (2u << 30); // [56:32]+type=2

    i32x8 g1;
    // data_size=4B (code 2), pad_enable, pad_interval=32dw, pad_amount=1dw
    g1[0] = (int)((2u << 16) | (1u << 20) | (4u << 22));
    g1[1] = (int)((tile_w & 0xFFFFu) << 16);                    // tensor_dim0 lo
    g1[2] = (int)((tile_w >> 16) | ((tensor_h & 0xFFFFu) << 16)); // dim0 hi|dim1 lo
    g1[3] = (int)((tensor_h >> 16) | (tile_w << 16));           // dim1 hi|tile_dim0
    g1[4] = (int)(tile_h & 0xFFFFu);                            // tile_dim1 (dim2=0)
    g1[5] = (int)row_stride;                                    // dim0_stride lo32
    g1[6] = 0;                                                  // stride hi / dim1_stride
    g1[7] = 0;
    i32x4 gz = {0, 0, 0, 0};
#if defined(__clang_major__) && (__clang_major__ >= 23)
    i32x8 gz8 = {0, 0, 0, 0, 0, 0, 0, 0};
    __builtin_amdgcn_tensor_load_to_lds(g0, g1, gz, gz, gz8, 0);
#else
    __builtin_amdgcn_tensor_load_to_lds(g0, g1, gz, gz, 0);
#endif
}
#endif

__global__ __launch_bounds__(256) void gemm_bias_kernel(
    const float* __restrict__ A, const float* __restrict__ W,
    const float* __restrict__ bias, float* __restrict__ Y,
    int M, int K, int N, int do_relu)
{
    __shared__ float As[2][TM][TK + 1];
    __shared__ float Ws[2][TK][TN + 1];

    const int tid   = threadIdx.x;
    const int lane  = tid & 31;
    const int wave  = tid >> 5;        // 0..7
    const int wm    = wave & 3;        // M sub-tile 0..3
    const int wn    = wave >> 2;       // N sub-tile 0..1
    const int r     = lane & 15;       // row/col within 16
    const int khalf = lane >> 4;       // lane>=16 owns the K+2 / M+8 half

    const int rowBase = blockIdx.x * TM;
    const int colBase = blockIdx.y * TN;

    v8f acc = {};

#if USE_TDM
    const bool issuer = (wave == 0);   // one wave drives the TDM
    const unsigned int remRows = (unsigned int)(M - rowBase);
    if (issuer) {
        tdm_load_2d((unsigned int)(uintptr_t)&As[0][0][0],
                    A + (size_t)rowBase * K, TK, TM, remRows, (unsigned)K);
        tdm_load_2d((unsigned int)(uintptr_t)&Ws[0][0][0],
                    W + colBase, TN, TK, TK, (unsigned)N);
    }
    int buf = 0;
    for (int kc = 0; kc < K; kc += TK, buf ^= 1) {
        if (issuer) {
            if (kc + TK < K) {   // prefetch next chunk into the other buffer
                tdm_load_2d((unsigned int)(uintptr_t)&As[buf ^ 1][0][0],
                            A + (size_t)rowBase * K + (kc + TK),
                            TK, TM, remRows, (unsigned)K);
                tdm_load_2d((unsigned int)(uintptr_t)&Ws[buf ^ 1][0][0],
                            W + (size_t)(kc + TK) * N + colBase,
                            TN, TK, TK, (unsigned)N);
                // in-order completion: <=2 outstanding => current pair landed
                __builtin_amdgcn_s_wait_tensorcnt(2);
            } else {
                __builtin_amdgcn_s_wait_tensorcnt(0);
            }
        }
        __syncthreads();                       // LDS tiles visible to all waves

        #pragma unroll
        for (int k0 = 0; k0 < TK; k0 += 4) {   // 8 x V_WMMA_F32_16X16X4_F32
            int kk = k0 + khalf * 2;
            v2f a, b;
            a.x = As[buf][wm * 16 + r][kk + 0];
            a.y = As[buf][wm * 16 + r][kk + 1];
            b.x = Ws[buf][kk + 0][wn * 16 + r];
            b.y = Ws[buf][kk + 1][wn * 16 + r];
            acc = __builtin_amdgcn_wmma_f32_16x16x4_f32(
                      false, a, false, b, (short)0, acc, false, false);
        }
        __syncthreads();                       // done reading before overwrite
    }
#else
    // Fallback: cooperative 128-bit staging loads (no TDM builtin available)
    for (int kc = 0; kc < K; kc += TK) {
        int ar = tid >> 3;
        int ac = (tid & 7) * 4;
        #pragma unroll
        for (int half = 0; half < 2; ++half) {
            int rr = ar + half * 32;
            int grow = rowBase + rr;
            float4 v = make_float4(0.f, 0.f, 0.f, 0.f);
            if (grow < M)
                v = *reinterpret_cast<const float4*>(&A[(size_t)grow * K + kc + ac]);
            As[0][rr][ac + 0] = v.x; As[0][rr][ac + 1] = v.y;
            As[0][rr][ac + 2] = v.z; As[0][rr][ac + 3] = v.w;
        }
        float4 v = *reinterpret_cast<const float4*>(
                       &W[(size_t)(kc + ar) * N + colBase + ac]);
        Ws[0][ar][ac + 0] = v.x; Ws[0][ar][ac + 1] = v.y;
        Ws[0][ar][ac + 2] = v.z; Ws[0][ar][ac + 3] = v.w;
        __syncthreads();
        if (kc + TK < K) {
            int grow = rowBase + (tid >> 3);
            if (grow < M)
                __builtin_prefetch(&A[(size_t)grow * K + kc + TK], 0, 0);
        }
        #pragma unroll
        for (int k0 = 0; k0 < TK; k0 += 4) {
            int kk = k0 + khalf * 2;
            v2f a, b;
            a.x = As[0][wm * 16 + r][kk + 0];
            a.y = As[0][wm * 16 + r][kk + 1];
            b.x = Ws[0][kk + 0][wn * 16 + r];
            b.y = Ws[0][kk + 1][wn * 16 + r];
            acc = __builtin_amdgcn_wmma_f32_16x16x4_f32(
                      false, a, false, b, (short)0, acc, false, false);
        }
        __syncthreads();
    }
#endif

    // store: C/D vgpr v -> M = v (+8 for lanes>=16), N = lane&15
    const int col = colBase + wn * 16 + r;
    #pragma unroll
    for (int v = 0; v < 8; ++v) {
        int row = rowBase + wm * 16 + v + khalf * 8;
        if (row < M) {
            float out = acc[v] + bias[col];
            if (do_relu) out = fmaxf(out, 0.f);
            Y[(size_t)row * N + col] = out;
        }
    }
}

// ---------------------------------------------------------------------------
// 3) Softmax over contiguous groups of 16 (per head: NLVL*NP logits)
// ---------------------------------------------------------------------------
__global__ __launch_bounds__(256) void softmax16_kernel(float* __restrict__ a,
                                                        int ngroups)
{
    int i = blockIdx.x * 256 + threadIdx.x;
    if (i >= ngroups) return;
    float* p = a + (size_t)i * 16;
    float mx = p[0];
    #pragma unroll
    for (int j = 1; j < 16; ++j) mx = fmaxf(mx, p[j]);
    float s = 0.f, e[16];
    #pragma unroll
    for (int j = 0; j < 16; ++j) { e[j] = __expf(p[j] - mx); s += e[j]; }
    float inv = 1.f / s;
    #pragma unroll
    for (int j = 0; j < 16; ++j) p[j] = e[j] * inv;
}

// ---------------------------------------------------------------------------
// 4) Deformable sampling: one wave per (b,q,head); lane = channel (HD==32).
// ---------------------------------------------------------------------------
__global__ __launch_bounds__(256) void msdeform_sample_kernel(
    const float* __restrict__ val,   // [B*Q, 256] == [B,Q,NH,HD]
    const float* __restrict__ off,   // [B*Q, 256] == [.,NH,NLVL,NP,2]
    const float* __restrict__ attn,  // [B*Q, 128] softmaxed
    float* __restrict__ out)         // [B*Q, 256]
{
    const int lane = threadIdx.x & 31;
    const int g = blockIdx.x * 8 + (threadIdx.x >> 5);
    if (g >= B_ * Q_ * NH_) return;
    const int h = g % NH_;
    const int q = (g / NH_) % Q_;
    const int b = g / (NH_ * Q_);
    const int m = b * Q_ + q;

    int stq, wq;
    if (q < 10000)      { stq = 0;     wq = 100; }
    else if (q < 12500) { stq = 10000; wq = 50;  }
    else if (q < 13125) { stq = 12500; wq = 25;  }
    else                { stq = 13125; wq = 13;  }
    int pq = q - stq;
    float refx = ((pq % wq) + 0.5f) / (float)wq;
    float refy = ((pq / wq) + 0.5f) / (float)wq;   // H == W for all levels

    const float* offm = off  + (size_t)m * 256 + h * 32;
    const float* am   = attn + (size_t)m * 128 + h * 16;

    const int starts[4] = {0, 10000, 12500, 13125};
    const int dims[4]   = {100, 50, 25, 13};

    float acc = 0.f;
    #pragma unroll
    for (int l = 0; l < NLVL_; ++l) {
        const int Wl = dims[l], Hl = dims[l];
        const float* vbase =
            val + (size_t)(b * Q_ + starts[l]) * 256 + h * 32 + lane;
        #pragma unroll
        for (int p = 0; p < NP_; ++p) {
            float ox = offm[(l * NP_ + p) * 2 + 0];
            float oy = offm[(l * NP_ + p) * 2 + 1];
            float a  = am[l * NP_ + p];
            float x = (refx + ox / (float)Wl) * (float)Wl - 0.5f;
            float y = (refy + oy / (float)Hl) * (float)Hl - 0.5f;
            float x0f = floorf(x), y0f = floorf(y);
            float fx = x - x0f, fy = y - y0f;
            int x0 = (int)x0f, y0 = (int)y0f;
            float s = 0.f;
            #pragma unroll
            for (int dy = 0; dy < 2; ++dy) {
                #pragma unroll
                for (int dx = 0; dx < 2; ++dx) {
                    int xi = x0 + dx, yi = y0 + dy;
                    if (xi >= 0 && xi < Wl && yi >= 0 && yi < Hl) {
                        float w = (dx ? fx : 1.f - fx) * (dy ? fy : 1.f - fy);
                        s += w * vbase[(size_t)(yi * Wl + xi) * 256];
                    }
                }
            }
            acc += a * s;
        }
    }
    out[(size_t)m * 256 + h * 32 + lane] = acc;
}

// ---------------------------------------------------------------------------
// 5) Fused residual + LayerNorm, in place: x = LN(x + delta) * g + b
// ---------------------------------------------------------------------------
__global__ __launch_bounds__(256) void add_ln_kernel(
    float* __restrict__ x, const float* __restrict__ delta,
    const float* __restrict__ g, const float* __restrict__ bta, int Mtot)
{
    const int lane = threadIdx.x & 31;
    const int row = blockIdx.x * 8 + (threadIdx.x >> 5);
    if (row >= Mtot) return;
    float* xr = x + (size_t)row * 256;
    const float* dr = delta + (size_t)row * 256;

    float v[8];
    float s = 0.f;
    #pragma unroll
    for (int j = 0; j < 8; ++j) {
        int d = lane + j * 32;
        v[j] = xr[d] + dr[d];
        s += v[j];
    }
    #pragma unroll
    for (int o = 16; o > 0; o >>= 1) s += __shfl_xor(s, o, 32);
    float mean = s * (1.f / 256.f);
    float vs = 0.f;
    #pragma unroll
    for (int j = 0; j < 8; ++j) { float t = v[j] - mean; vs += t * t; }
    #pragma unroll
    for (int o = 16; o > 0; o >>= 1) vs += __shfl_xor(vs, o, 32);
    float inv = rsqrtf(vs * (1.f / 256.f) + 1e-5f);
    #pragma unroll
    for (int j = 0; j < 8; ++j) {
        int d = lane + j * 32;
        xr[d] = (v[j] - mean) * inv * g[d] + bta[d];
    }
}

// ---------------------------------------------------------------------------
// Host-side orchestration
// ---------------------------------------------------------------------------
static inline void gemm(const float* A, const float* W, const float* bias,
                        float* Y, int M, int K, int N, int relu,
                        hipStream_t s)
{
    dim3 grid((M + TM - 1) / TM, N / TN);
    gemm_bias_kernel<<<grid, 256, 0, s>>>(A, W, bias, Y, M, K, N, relu);
}

extern "C" void kernel_launch(void* const* d_in, const int* in_sizes, int n_in,
                              void* d_out, int out_size, void* d_ws,
                              size_t ws_size, hipStream_t stream)
{
    (void)in_sizes; (void)n_in; (void)ws_size;
    const float* src[4] = {(const float*)d_in[0], (const float*)d_in[1],
                           (const float*)d_in[2], (const float*)d_in[3]};
    const float* pos[4] = {(const float*)d_in[4], (const float*)d_in[5],
                           (const float*)d_in[6], (const float*)d_in[7]};
    const float* lev   = (const float*)d_in[8];
    const float* Woff  = (const float*)d_in[9];
    const float* boff  = (const float*)d_in[10];
    const float* Wattn = (const float*)d_in[11];
    const float* battn = (const float*)d_in[12];
    const float* Wv    = (const float*)d_in[13];
    const float* bv    = (const float*)d_in[14];
    const float* Wo    = (const float*)d_in[15];
    const float* bo    = (const float*)d_in[16];
    const float* g1    = (const float*)d_in[17];
    const float* b1    = (const float*)d_in[18];
    const float* W1    = (const float*)d_in[19];
    const float* bb1   = (const float*)d_in[20];
    const float* W2    = (const float*)d_in[21];
    const float* bb2   = (const float*)d_in[22];
    const float* g2    = (const float*)d_in[23];
    const float* b2    = (const float*)d_in[24];

    // workspace partition (floats); total = M*(7*256 + 128 + 1024) ~= 313MB
    float* ws   = (float*)d_ws;
    const size_t MD = (size_t)M_ * 256;
    float* X    = ws;
    float* POS  = X    + MD;
    float* QB   = POS  + MD;
    float* OFF  = QB   + MD;
    float* ATT  = OFF  + MD;
    float* VAL  = ATT  + (size_t)M_ * 128;
    float* SAMP = VAL  + MD;
    float* TMP  = SAMP + MD;
    float* HID  = TMP  + MD;

    const int totBQD = B_ * Q_ * D_;
    flatten_kernel<<<(totBQD + 255) / 256, 256, 0, stream>>>(
        src[0], src[1], src[2], src[3], pos[0], pos[1], pos[2], pos[3],
        lev, X, POS);

    const int n4   = (int)(MD / 4);
    const int lnBl = (M_ + 7) / 8;
    const int smBl = (M_ * NH_ + 255) / 256;
    const int spBl = (B_ * Q_ * NH_ + 7) / 8;

    for (int l = 0; l < NLAYERS_; ++l) {
        const float* Woff_l  = Woff  + (size_t)l * 256 * 256;
        const float* boff_l  = boff  + (size_t)l * 256;
        const float* Wattn_l = Wattn + (size_t)l * 256 * 128;
        const float* battn_l = battn + (size_t)l * 128;
        const float* Wv_l    = Wv    + (size_t)l * 256 * 256;
        const float* bv_l    = bv    + (size_t)l * 256;
        const float* Wo_l    = Wo    + (size_t)l * 256 * 256;
        const float* bo_l    = bo    + (size_t)l * 256;
        const float* W1_l    = W1    + (size_t)l * 256 * 1024;
        const float* bb1_l   = bb1   + (size_t)l * 1024;
        const float* W2_l    = W2    + (size_t)l * 1024 * 256;
        const float* bb2_l   = bb2   + (size_t)l * 256;

        // q = x + pos
        add4_kernel<<<(n4 + 255) / 256, 256, 0, stream>>>(X, POS, QB, n4);
        // projections (WMMA fp32, TDM-staged tiles)
        gemm(QB, Woff_l, boff_l, OFF, M_, 256, 256, 0, stream);
        gemm(QB, Wattn_l, battn_l, ATT, M_, 256, 128, 0, stream);
        softmax16_kernel<<<smBl, 256, 0, stream>>>(ATT, M_ * NH_);
        gemm(X, Wv_l, bv_l, VAL, M_, 256, 256, 0, stream);
        // deformable gather-attention
        msdeform_sample_kernel<<<spBl, 256, 0, stream>>>(VAL, OFF, ATT, SAMP);
        // output proj + residual + LN1
        gemm(SAMP, Wo_l, bo_l, TMP, M_, 256, 256, 0, stream);
        add_ln_kernel<<<lnBl, 256, 0, stream>>>(X, TMP,
                                                g1 + (size_t)l * 256,
                                                b1 + (size_t)l * 256, M_);
        // FFN + residual + LN2
        gemm(X, W1_l, bb1_l, HID, M_, 256, 1024, 1, stream);
        gemm(HID, W2_l, bb2_l, TMP, M_, 1024, 256, 0, stream);
        add_ln_kernel<<<lnBl, 256, 0, stream>>>(X, TMP,
                                                g2 + (size_t)l * 256,
                                                b2 + (size_t)l * 256, M_);
    }

    hipMemcpyAsync(d_out, X, (size_t)out_size * sizeof(float),
                   hipMemcpyDeviceToDevice, stream);
}